// GAT_Net_76776835383826
// MI455X (gfx1250) — compile-verified
//
#include <hip/hip_runtime.h>
#include <math.h>

#define N_NODES 100000
#define N_EDGES 1600000
#define E_TOT   (N_EDGES + N_NODES)   // self-loops appended
#define IN_DIM  128
#define HID     32
#define HEADS   4
#define HC      (HEADS * HID)         // 128
#define NCLS    2

typedef __attribute__((ext_vector_type(2))) float v2f;
typedef __attribute__((ext_vector_type(8))) float v8f;

__device__ __forceinline__ float leakyf(float v, float s) { return v > 0.f ? v : s * v; }

// float atomic max via signed-max / unsigned-min bit trick (init value must be -inf)
__device__ __forceinline__ void atomicMaxF32(float* addr, float val) {
  if (val >= 0.f) atomicMax((int*)addr, __float_as_int(val));
  else            atomicMin((unsigned int*)addr, __float_as_uint(val));
}

// ---------------------------------------------------------------- init
__global__ void k_init(float* __restrict__ num1, float* __restrict__ m1, float* __restrict__ den1,
                       float* __restrict__ num2, float* __restrict__ m2, float* __restrict__ den2) {
  int i = blockIdx.x * blockDim.x + threadIdx.x;
  if (i < N_NODES * HC)    num1[i] = 0.f;
  if (i < N_NODES * HEADS) { m1[i] = -INFINITY; den1[i] = 0.f; }
  if (i < N_NODES * NCLS)  num2[i] = 0.f;
  if (i < N_NODES)         { m2[i] = -INFINITY; den2[i] = 0.f; }
}

// ---------------------------------------------------------------- layer-1 GEMM: h1 = x @ W1^T
// Wave32 per 16x16 tile, K=128 via 32 chained V_WMMA_F32_16X16X4_F32.
// Block = 256 threads = 8 waves; wave w owns output columns [16w,16w+16); block b owns rows [16b,16b+16).
__global__ void __launch_bounds__(256)
k_gemm1_wmma(const float* __restrict__ X, const float* __restrict__ W, float* __restrict__ H) {
  const int node0 = blockIdx.x * 16;
  const int wave  = threadIdx.x >> 5;
  const int lane  = threadIdx.x & 31;
  const int out0  = wave * 16;
  const int half  = lane >> 4;       // 0: K pair {0,1}; 1: K pair {2,3}
  const int l16   = lane & 15;

  const float* __restrict__ xrow = X + (node0 + l16) * IN_DIM + half * 2; // A: M=l16
  const float* __restrict__ wrow = W + (out0  + l16) * IN_DIM + half * 2; // B=W1^T: N=l16

  v8f acc = {};
#pragma unroll
  for (int k = 0; k < IN_DIM; k += 4) {
    v2f a, b;
    a.x = xrow[k]; a.y = xrow[k + 1];
    b.x = wrow[k]; b.y = wrow[k + 1];
    acc = __builtin_amdgcn_wmma_f32_16x16x4_f32(false, a, false, b, (short)0, acc,
                                                false, false);
  }
  // C/D layout: VGPR j -> M=j (lanes 0-15) / M=8+j (lanes 16-31), N = l16
#pragma unroll
  for (int j = 0; j < 8; ++j) {
    int row = node0 + half * 8 + j;
    H[row * HC + out0 + l16] = acc[j];
  }
}

// ---------------------------------------------------------------- layer-1 attention dots
__global__ void k_attdot1(const float* __restrict__ h1, const float* __restrict__ asw,
                          const float* __restrict__ adw,
                          float* __restrict__ as1, float* __restrict__ ad1) {
  int i = blockIdx.x * blockDim.x + threadIdx.x;   // i = n*HEADS + h
  if (i >= N_NODES * HEADS) return;
  int h = i & (HEADS - 1);
  int n = i >> 2;
  const float* hp = h1 + n * HC + h * HID;
  const float* sp = asw + h * HID;
  const float* dp = adw + h * HID;
  float s = 0.f, d = 0.f;
#pragma unroll
  for (int c = 0; c < HID; ++c) { float v = hp[c]; s = fmaf(v, sp[c], s); d = fmaf(v, dp[c], d); }
  as1[i] = s; ad1[i] = d;
}

// ---------------------------------------------------------------- layer-1 edge pass 1: segment max
__global__ void k_edgemax1(const long long* __restrict__ ei, const float* __restrict__ as1,
                           const float* __restrict__ ad1, float* __restrict__ m1) {
  long long i = (long long)blockIdx.x * blockDim.x + threadIdx.x;  // i = e*HEADS + h
  if (i >= (long long)E_TOT * HEADS) return;
  int e = (int)(i >> 2), h = (int)(i & 3);
  int src, dst;
  if (e < N_EDGES) { src = (int)ei[e]; dst = (int)ei[N_EDGES + e]; }
  else             { src = dst = e - N_EDGES; }
  float ee = leakyf(as1[src * HEADS + h] + ad1[dst * HEADS + h], 0.2f);
  atomicMaxF32(&m1[dst * HEADS + h], ee);
}

__global__ void k_fixm(float* __restrict__ m, int len) {
  int i = blockIdx.x * blockDim.x + threadIdx.x;
  if (i >= len) return;
  float v = m[i];
  if (!isfinite(v)) m[i] = 0.f;
}

// ---------------------------------------------------------------- layer-1 edge pass 2: fused num/denom
__global__ void k_edgeacc1(const long long* __restrict__ ei, const float* __restrict__ as1,
                           const float* __restrict__ ad1, const float* __restrict__ m1,
                           const float* __restrict__ h1,
                           float* __restrict__ den1, float* __restrict__ num1) {
  long long i = (long long)blockIdx.x * blockDim.x + threadIdx.x;  // i = e*HEADS + h
  if (i >= (long long)E_TOT * HEADS) return;
  int e = (int)(i >> 2), h = (int)(i & 3);
  int src, dst;
  if (e < N_EDGES) { src = (int)ei[e]; dst = (int)ei[N_EDGES + e]; }
  else             { src = dst = e - N_EDGES; }
  float ee = leakyf(as1[src * HEADS + h] + ad1[dst * HEADS + h], 0.2f);
  float ex = expf(ee - m1[dst * HEADS + h]);
  atomicAdd(&den1[dst * HEADS + h], ex);
  const float* hp = h1 + src * HC + h * HID;
  float* np = num1 + dst * HC + h * HID;
#pragma unroll
  for (int c = 0; c < HID; ++c) atomicAdd(&np[c], ex * hp[c]);
}

// ---------------------------------------------------------------- layer-1 finalize (in-place num1 -> activations)
__global__ void k_final1(const float* __restrict__ den1, const float* __restrict__ b1,
                         float* __restrict__ num1) {
  int i = blockIdx.x * blockDim.x + threadIdx.x;   // i = n*HC + c
  if (i >= N_NODES * HC) return;
  int n = i >> 7, c = i & 127, h = c >> 5;
  float v = num1[i] / (den1[n * HEADS + h] + 1e-16f) + b1[c];
  num1[i] = leakyf(v, 0.01f);   // F.leaky_relu default slope
}

// ---------------------------------------------------------------- layer-2 projection + attention dots
__global__ void k_proj2(const float* __restrict__ hact, const float* __restrict__ W2,
                        const float* __restrict__ asw, const float* __restrict__ adw,
                        float* __restrict__ h2, float* __restrict__ as2, float* __restrict__ ad2) {
  int n = blockIdx.x * blockDim.x + threadIdx.x;
  if (n >= N_NODES) return;
  const float4* hp = (const float4*)(hact + n * HC);
  const float4* w0 = (const float4*)W2;
  const float4* w1 = (const float4*)(W2 + HC);
  float z0 = 0.f, z1 = 0.f;
#pragma unroll
  for (int k = 0; k < HC / 4; ++k) {
    float4 hv = hp[k], a = w0[k], b = w1[k];
    z0 = fmaf(hv.x, a.x, fmaf(hv.y, a.y, fmaf(hv.z, a.z, fmaf(hv.w, a.w, z0))));
    z1 = fmaf(hv.x, b.x, fmaf(hv.y, b.y, fmaf(hv.z, b.z, fmaf(hv.w, b.w, z1))));
  }
  h2[n * 2] = z0; h2[n * 2 + 1] = z1;
  as2[n] = z0 * asw[0] + z1 * asw[1];
  ad2[n] = z0 * adw[0] + z1 * adw[1];
}

// ---------------------------------------------------------------- layer-2 edge passes
__global__ void k_edgemax2(const long long* __restrict__ ei, const float* __restrict__ as2,
                           const float* __restrict__ ad2, float* __restrict__ m2) {
  int e = blockIdx.x * blockDim.x + threadIdx.x;
  if (e >= E_TOT) return;
  int src, dst;
  if (e < N_EDGES) { src = (int)ei[e]; dst = (int)ei[N_EDGES + e]; }
  else             { src = dst = e - N_EDGES; }
  float ee = leakyf(as2[src] + ad2[dst], 0.2f);
  atomicMaxF32(&m2[dst], ee);
}

__global__ void k_edgeacc2(const long long* __restrict__ ei, const float* __restrict__ as2,
                           const float* __restrict__ ad2, const float* __restrict__ m2,
                           const float* __restrict__ h2,
                           float* __restrict__ den2, float* __restrict__ num2) {
  int e = blockIdx.x * blockDim.x + threadIdx.x;
  if (e >= E_TOT) return;
  int src, dst;
  if (e < N_EDGES) { src = (int)ei[e]; dst = (int)ei[N_EDGES + e]; }
  else             { src = dst = e - N_EDGES; }
  float ee = leakyf(as2[src] + ad2[dst], 0.2f);
  float ex = expf(ee - m2[dst]);
  atomicAdd(&den2[dst], ex);
  atomicAdd(&num2[dst * 2 + 0], ex * h2[src * 2 + 0]);
  atomicAdd(&num2[dst * 2 + 1], ex * h2[src * 2 + 1]);
}

// ---------------------------------------------------------------- finalize + log_softmax
__global__ void k_final2(const float* __restrict__ num2, const float* __restrict__ den2,
                         const float* __restrict__ b2, float* __restrict__ out) {
  int n = blockIdx.x * blockDim.x + threadIdx.x;
  if (n >= N_NODES) return;
  float inv = 1.f / (den2[n] + 1e-16f);
  float o0 = num2[n * 2 + 0] * inv + b2[0];
  float o1 = num2[n * 2 + 1] * inv + b2[1];
  float mx = fmaxf(o0, o1);
  float l0 = o0 - mx, l1 = o1 - mx;
  float lse = logf(expf(l0) + expf(l1));
  out[n * 2 + 0] = l0 - lse;
  out[n * 2 + 1] = l1 - lse;
}

// ================================================================ launch
extern "C" void kernel_launch(void* const* d_in, const int* in_sizes, int n_in,
                              void* d_out, int out_size, void* d_ws, size_t ws_size,
                              hipStream_t stream) {
  const float*     x    = (const float*)d_in[0];
  const long long* ei   = (const long long*)d_in[1];   // int64 [2, N_EDGES]
  const float*     W1   = (const float*)d_in[2];
  const float*     as1w = (const float*)d_in[3];
  const float*     ad1w = (const float*)d_in[4];
  const float*     b1   = (const float*)d_in[5];
  const float*     W2   = (const float*)d_in[6];
  const float*     as2w = (const float*)d_in[7];
  const float*     ad2w = (const float*)d_in[8];
  const float*     b2   = (const float*)d_in[9];
  float*           out  = (float*)d_out;

  // workspace layout (floats)
  float* ws   = (float*)d_ws;
  float* h1   = ws;                       // N*128
  float* num1 = h1   + (size_t)N_NODES * HC;      // N*128 (becomes activations in-place)
  float* as1  = num1 + (size_t)N_NODES * HC;      // N*4
  float* ad1  = as1  + (size_t)N_NODES * HEADS;   // N*4
  float* m1   = ad1  + (size_t)N_NODES * HEADS;   // N*4
  float* den1 = m1   + (size_t)N_NODES * HEADS;   // N*4
  float* h2   = den1 + (size_t)N_NODES * HEADS;   // N*2
  float* num2 = h2   + (size_t)N_NODES * NCLS;    // N*2
  float* as2  = num2 + (size_t)N_NODES * NCLS;    // N
  float* ad2  = as2  + (size_t)N_NODES;           // N
  float* m2   = ad2  + (size_t)N_NODES;           // N
  float* den2 = m2   + (size_t)N_NODES;           // N

  const int B = 256;
  const int gNHC  = (N_NODES * HC + B - 1) / B;          // 50000
  const int gNH   = (N_NODES * HEADS + B - 1) / B;       // 1563
  const int gEH   = (int)(((long long)E_TOT * HEADS + B - 1) / B); // 26563
  const int gE    = (E_TOT + B - 1) / B;                 // 6641
  const int gN    = (N_NODES + B - 1) / B;               // 391

  k_init<<<gNHC, B, 0, stream>>>(num1, m1, den1, num2, m2, den2);

  // layer 1
  k_gemm1_wmma<<<N_NODES / 16, 256, 0, stream>>>(x, W1, h1);
  k_attdot1<<<gNH, B, 0, stream>>>(h1, as1w, ad1w, as1, ad1);
  k_edgemax1<<<gEH, B, 0, stream>>>(ei, as1, ad1, m1);
  k_fixm<<<gNH, B, 0, stream>>>(m1, N_NODES * HEADS);
  k_edgeacc1<<<gEH, B, 0, stream>>>(ei, as1, ad1, m1, h1, den1, num1);
  k_final1<<<gNHC, B, 0, stream>>>(den1, b1, num1);      // num1 -> activations in place

  // layer 2
  k_proj2<<<gN, B, 0, stream>>>(num1, W2, as2w, ad2w, h2, as2, ad2);
  k_edgemax2<<<gE, B, 0, stream>>>(ei, as2, ad2, m2);
  k_fixm<<<gN, B, 0, stream>>>(m2, N_NODES);
  k_edgeacc2<<<gE, B, 0, stream>>>(ei, as2, ad2, m2, h2, den2, num2);
  k_final2<<<gN, B, 0, stream>>>(num2, den2, b2, out);
}